// SAGELayer_66726611911054
// MI455X (gfx1250) — compile-verified
//
#include <hip/hip_runtime.h>

typedef __attribute__((ext_vector_type(2))) float v2f;
typedef __attribute__((ext_vector_type(8))) float v8f;

#define D 128           // D_IN == D_OUT == 128
#define EDGES_PER_BLOCK 8
#define LDS_STRIDE 132  // 128 + 4 pad: fragment reads hit all 64 LDS banks

// ---------------------------------------------------------------------------
// Step 2: edge-parallel gather + scatter-add (mean-aggregator numerator) and
// degree count. One wave32 per edge. feat row = 512B -> one float4 per lane,
// fully coalesced global_load_b128. Atomics resolve in L2 (feat 51MB +
// accumulator 51MB both fit the 192MB L2).
// ---------------------------------------------------------------------------
__global__ void sage_scatter(const float* __restrict__ feat,
                             const int* __restrict__ src,
                             const int* __restrict__ dst,
                             float* __restrict__ neigh,   // [N, D] accumulator (d_out)
                             float* __restrict__ deg,     // [N]    degree (float)
                             int E) {
    const int lane = threadIdx.x & 31;
    const int wave = threadIdx.x >> 5;
    const int e = blockIdx.x * EDGES_PER_BLOCK + wave;
    if (e >= E) return;

    const int s = src[e];
    const int d = dst[e];

    const float4* __restrict__ fp = (const float4*)(feat + (size_t)s * D);
    float4 v = fp[lane];                       // global_load_b128, coalesced

    float* np = neigh + (size_t)d * D + (size_t)lane * 4;
    unsafeAtomicAdd(np + 0, v.x);              // global_atomic_add_f32 x4
    unsafeAtomicAdd(np + 1, v.y);
    unsafeAtomicAdd(np + 2, v.z);
    unsafeAtomicAdd(np + 3, v.w);

    if (lane == 0) unsafeAtomicAdd(deg + d, 1.0f);
}

// ---------------------------------------------------------------------------
// Step 3: fused  out = feat @ Wself^T + b_self + (neigh/max(deg,1)) @ Wneigh^T
//
// Block = 256 threads = 8 wave32s; block owns a 16-row tile, wave w owns the
// 16-column tile [16w, 16w+16). The two A-tiles (feat row-tile and the
// degree-scaled neigh row-tile) are staged into LDS ONCE per block
// (coalesced float4 loads, rdeg folded in), then every wave feeds its
// V_WMMA_F32_16X16X4_F32 A-fragments from LDS (ds_load_b64) -- this removes
// the 8x redundant global A traffic. B fragments (weights, 128KB total,
// L2-hot) load directly from global.
//
// f32 16x4 A fragment (ISA 7.12.2): lane L -> M = L%16, holds K = 2*(L/16)
// and K = 2*(L/16)+1 -> one 8B-aligned v2f load. B (4x16) mirrors it with
// N = L%16. C/D: VGPR j -> row M = j + 8*(L/16).
//
// In-place safety: all global reads of `out` (neigh sums) happen in the
// staging phase before __syncthreads(); compute reads LDS only; blocks touch
// disjoint row tiles. Race-free with a single barrier.
// ---------------------------------------------------------------------------
__global__ void sage_gemm(const float* __restrict__ feat,
                          float* __restrict__ out,       // neigh-sum in, result out
                          const float* __restrict__ deg,
                          const float* __restrict__ Wself,   // [D_OUT, D_IN]
                          const float* __restrict__ Wneigh,  // [D_OUT, D_IN]
                          const float* __restrict__ bself,   // [D_OUT]
                          int N) {
    __shared__ float ls_feat [16][LDS_STRIDE];
    __shared__ float ls_neigh[16][LDS_STRIDE];

    const int row0 = blockIdx.x * 16;

    // ---- Stage: 16 rows x 128 cols = 512 float4 per tile; 256 threads x 2.
#pragma unroll
    for (int i = 0; i < 2; ++i) {
        const int f4  = i * 256 + threadIdx.x;   // 0..511, lanes contiguous
        const int r   = f4 >> 5;                 // row within tile (0..15)
        const int c4  = f4 & 31;                 // float4 column (0..31)
        int row = row0 + r;
        if (row >= N) row = N - 1;               // clamp; stores guarded below

        const float4 f  = *(const float4*)(feat + (size_t)row * D + c4 * 4);
        const float4 ns = *(const float4*)(out  + (size_t)row * D + c4 * 4);
        const float rdeg = 1.0f / fmaxf(deg[row], 1.0f);

        float* pf = &ls_feat [r][c4 * 4];
        float* pn = &ls_neigh[r][c4 * 4];
        pf[0] = f.x;  pf[1] = f.y;  pf[2] = f.z;  pf[3] = f.w;
        pn[0] = ns.x * rdeg; pn[1] = ns.y * rdeg;
        pn[2] = ns.z * rdeg; pn[3] = ns.w * rdeg;
    }

    __syncthreads();   // staging (incl. all global reads of `out`) complete

    // ---- Compute: per-wave 16x16 tile via 64 chained f32 WMMAs.
    const int lane = threadIdx.x & 31;
    const int wave = threadIdx.x >> 5;     // 0..7 -> column tile
    const int col0 = wave * 16;
    const int mn   = lane & 15;            // M for A, N for B/C/D
    const int half = lane >> 4;            // selects K pair within each 4-step

    const float* aS = &ls_feat [mn][0];
    const float* aN = &ls_neigh[mn][0];
    const float* wsRow = Wself  + (size_t)(col0 + mn) * D;  // B[k][n] = W[n][k]
    const float* wnRow = Wneigh + (size_t)(col0 + mn) * D;

    v8f acc = {};
#pragma unroll
    for (int k0 = 0; k0 < D; k0 += 4) {
        const int k = k0 + 2 * half;
        v2f a_s = *(const v2f*)(aS + k);         // ds_load_b64, conflict-free
        v2f a_n = *(const v2f*)(aN + k);
        v2f b_s = *(const v2f*)(wsRow + k);      // global_load_b64, L2-hot
        v2f b_n = *(const v2f*)(wnRow + k);

        acc = __builtin_amdgcn_wmma_f32_16x16x4_f32(
            /*neg_a=*/false, a_s, /*neg_b=*/false, b_s,
            /*c_mod=*/(short)0, acc, /*reuse_a=*/false, /*reuse_b=*/false);
        acc = __builtin_amdgcn_wmma_f32_16x16x4_f32(
            false, a_n, false, b_n, (short)0, acc, false, false);
    }

    const float bias = bself[col0 + mn];

#pragma unroll
    for (int j = 0; j < 8; ++j) {
        const int row = row0 + j + 8 * half;     // C/D layout: M = j + 8*half
        if (row < N) out[(size_t)row * D + col0 + mn] = acc[j] + bias;
    }
}

// ---------------------------------------------------------------------------
// Launcher. Inputs (setup_inputs order):
//   0: feat   [N,128] f32      1: W_neigh [128,128] f32
//   2: W_self [128,128] f32    3: b_self  [128] f32
//   4: src    [E] i32          5: dst     [E] i32
// d_out: [N,128] f32 (also serves as the neigh-sum accumulator).
// d_ws: degree array (N floats).
// ---------------------------------------------------------------------------
extern "C" void kernel_launch(void* const* d_in, const int* in_sizes, int n_in,
                              void* d_out, int out_size, void* d_ws, size_t ws_size,
                              hipStream_t stream) {
    const float* feat   = (const float*)d_in[0];
    const float* Wneigh = (const float*)d_in[1];
    const float* Wself  = (const float*)d_in[2];
    const float* bself  = (const float*)d_in[3];
    const int*   src    = (const int*)d_in[4];
    const int*   dst    = (const int*)d_in[5];

    const int N = in_sizes[0] / D;
    const int E = in_sizes[4];

    float* out = (float*)d_out;
    float* deg = (float*)d_ws;

    // Zero the neigh-sum accumulator (in d_out) and the degree array.
    hipMemsetAsync(d_out, 0, (size_t)N * D * sizeof(float), stream);
    hipMemsetAsync(d_ws, 0, (size_t)N * sizeof(float), stream);

    const int scatterBlocks = (E + EDGES_PER_BLOCK - 1) / EDGES_PER_BLOCK;
    sage_scatter<<<scatterBlocks, 256, 0, stream>>>(feat, src, dst, out, deg, E);

    const int gemmBlocks = (N + 15) / 16;
    sage_gemm<<<gemmBlocks, 256, 0, stream>>>(feat, out, deg, Wself, Wneigh, bself, N);
}